// rsn_cluster_70437463654564
// MI455X (gfx1250) — compile-verified
//
#include <hip/hip_runtime.h>

// Problem constants (from reference setup_inputs)
#define B_   4
#define C_   2
#define H_   480
#define W_   640
#define P_   (H_ * W_)          // 307200 pixels
#define K_   32                 // seeds / means per batch
#define NIT  10
#define BW2  0.0256f            // BANDWIDTH^2 = 0.16^2
#define FAR  1.0e18f            // finite sentinel for inactive lanes

// Iteration pass partitioning: 64 blocks per batch, 4800 px each (exact)
#define NBLK   64
#define CHUNK  (P_ / NBLK)      // 4800
// Final pass partitioning: 150 blocks per batch, 2048 px each (exact)
#define NBLK2  150
#define CHUNK2 (P_ / NBLK2)     // 2048

typedef __attribute__((ext_vector_type(2))) float v2f;
typedef __attribute__((ext_vector_type(8))) float v8f;

// ---------------------------------------------------------------------------
// means0[b,k,:] = feature[b,:,seed_idx[b,k]]
// ---------------------------------------------------------------------------
__global__ void ms_init_means(const float* __restrict__ feat,
                              const int* __restrict__ seed,
                              float* __restrict__ means) {
    int t = threadIdx.x;
    if (t >= B_ * K_) return;
    int b = t / K_;
    int s = seed[t];
    const float* fb = feat + (size_t)b * C_ * P_;
    means[t * 2 + 0] = fb[s];
    means[t * 2 + 1] = fb[P_ + s];
}

// ---------------------------------------------------------------------------
// Build the two 16x4 A operands (means side) for one wave.
// Row m (lane j = m): K0=-2mx, K1=-2my (lanes 0-15) ; K2=|m|^2, K3=1 (16-31).
// ---------------------------------------------------------------------------
__device__ __forceinline__ void ms_make_A(const float* __restrict__ means,
                                          int b, int j, int hs,
                                          v2f& A0, v2f& A1) {
    float m0x = means[(b * K_ + j) * 2 + 0];
    float m0y = means[(b * K_ + j) * 2 + 1];
    float m1x = means[(b * K_ + 16 + j) * 2 + 0];
    float m1y = means[(b * K_ + 16 + j) * 2 + 1];
    A0.x = hs ? fmaf(m0x, m0x, m0y * m0y) : (-2.0f * m0x);
    A0.y = hs ? 1.0f                      : (-2.0f * m0y);
    A1.x = hs ? fmaf(m1x, m1x, m1y * m1y) : (-2.0f * m1x);
    A1.y = hs ? 1.0f                      : (-2.0f * m1y);
}

// ---------------------------------------------------------------------------
// Accumulation pass: per block, sum (count, sum_fx, sum_fy) per mean over its
// pixel chunk, using V_WMMA_F32_16X16X4_F32 squared-distance tiles.
// Deterministic: per-wave shuffle reduce -> per-wave LDS slots -> fixed-order
// cross-wave sum -> partials[b][blk][k][3].
// ---------------------------------------------------------------------------
__global__ void ms_accum(const float* __restrict__ feat,
                         const float* __restrict__ means,
                         float* __restrict__ partials) {
    __shared__ float red[8][K_][3];   // 8 waves per 256-thread block

    int bid  = blockIdx.x;
    int b    = bid / NBLK;
    int blk  = bid % NBLK;
    int tid  = threadIdx.x;
    int lane = tid & 31;
    int wave = tid >> 5;
    int j    = lane & 15;   // pixel slot within group / mean row
    int hs   = lane >> 4;   // half-select: K-pair for A/B operands

    const float* fb = feat + (size_t)b * C_ * P_;

    v2f A0, A1;
    ms_make_A(means, b, j, hs, A0, A1);

    float aN[2][8] = {}, aX[2][8] = {}, aY[2][8] = {};

    const int start  = blk * CHUNK;
    const int end    = start + CHUNK;
    const int sweeps = (CHUNK + 127) / 128;   // 38 (last one partial)

    for (int s = 0; s < sweeps; ++s) {
        int  p   = start + s * 128 + wave * 16 + j;
        bool act = p < end;
        float fx = act ? fb[p]       : FAR;
        float fy = act ? fb[P_ + p]  : FAR;

        v2f Bv;
        Bv.x = hs ? 1.0f                    : fx;
        Bv.y = hs ? fmaf(fx, fx, fy * fy)   : fy;

        v8f cz = {};
        // D[m,p] = -2 m.f + |m|^2 + |f|^2 = squared distance
        v8f d0 = __builtin_amdgcn_wmma_f32_16x16x4_f32(
            false, A0, false, Bv, (short)0, cz, false, false);
        v8f d1 = __builtin_amdgcn_wmma_f32_16x16x4_f32(
            false, A1, false, Bv, (short)0, cz, false, false);

#pragma unroll
        for (int r = 0; r < 8; ++r) {
            float m0 = (d0[r] < BW2) ? 1.0f : 0.0f;
            float m1 = (d1[r] < BW2) ? 1.0f : 0.0f;
            aN[0][r] += m0;
            aX[0][r]  = fmaf(m0, fx, aX[0][r]);
            aY[0][r]  = fmaf(m0, fy, aY[0][r]);
            aN[1][r] += m1;
            aX[1][r]  = fmaf(m1, fx, aX[1][r]);
            aY[1][r]  = fmaf(m1, fy, aY[1][r]);
        }
    }

    // Reduce across the 16 pixel-lanes of each half (xor 1,2,4,8 stays in-half)
#pragma unroll
    for (int t = 0; t < 2; ++t) {
#pragma unroll
        for (int r = 0; r < 8; ++r) {
            float n = aN[t][r], x = aX[t][r], y = aY[t][r];
#pragma unroll
            for (int m = 1; m <= 8; m <<= 1) {
                n += __shfl_xor(n, m, 32);
                x += __shfl_xor(x, m, 32);
                y += __shfl_xor(y, m, 32);
            }
            if (j == 0) {   // lanes 0 and 16 hold the two half-sums
                int k = t * 16 + r + hs * 8;
                red[wave][k][0] = n;
                red[wave][k][1] = x;
                red[wave][k][2] = y;
            }
        }
    }
    __syncthreads();

    if (tid < K_ * 3) {
        int k = tid / 3, c = tid % 3;
        float s = 0.0f;
#pragma unroll
        for (int w = 0; w < 8; ++w) s += red[w][k][c];
        partials[(((size_t)b * NBLK + blk) * K_ + k) * 3 + c] = s;
    }
}

// ---------------------------------------------------------------------------
// means[b,k] = sum(partials)/count, fixed summation order (deterministic)
// ---------------------------------------------------------------------------
__global__ void ms_update(const float* __restrict__ partials,
                          float* __restrict__ means) {
    int t = threadIdx.x;
    if (t >= B_ * K_) return;
    int b = t / K_, k = t % K_;
    float n = 0.0f, x = 0.0f, y = 0.0f;
    for (int blk = 0; blk < NBLK; ++blk) {
        const float* p = &partials[(((size_t)b * NBLK + blk) * K_ + k) * 3];
        n += p[0]; x += p[1]; y += p[2];
    }
    means[t * 2 + 0] = x / n;
    means[t * 2 + 1] = y / n;
}

__global__ void ms_write_means(const float* __restrict__ means,
                               float* __restrict__ out) {
    int t = threadIdx.x;
    if (t < B_ * K_ * C_) out[(size_t)B_ * P_ + t] = means[t];
}

// ---------------------------------------------------------------------------
// Final pass: same WMMA distance tiles; emit mask floats and first-k labels.
// Output layout (all f32): labels[B*P] | means[B*K*2] | mask[B*K*P]
// ---------------------------------------------------------------------------
__global__ void ms_final(const float* __restrict__ feat,
                         const float* __restrict__ means,
                         float* __restrict__ out) {
    int bid  = blockIdx.x;
    int b    = bid / NBLK2;
    int blk  = bid % NBLK2;
    int lane = threadIdx.x & 31;
    int wave = threadIdx.x >> 5;
    int j    = lane & 15;
    int hs   = lane >> 4;

    const float* fb = feat + (size_t)b * C_ * P_;
    float* labels = out;
    float* masko  = out + (size_t)B_ * P_ + (size_t)B_ * K_ * C_;

    v2f A0, A1;
    ms_make_A(means, b, j, hs, A0, A1);

    const int start  = blk * CHUNK2;
    const int sweeps = CHUNK2 / 128;   // 16, exact

    for (int s = 0; s < sweeps; ++s) {
        int p = start + s * 128 + wave * 16 + j;
        float fx = fb[p];
        float fy = fb[P_ + p];

        v2f Bv;
        Bv.x = hs ? 1.0f                  : fx;
        Bv.y = hs ? fmaf(fx, fx, fy * fy) : fy;

        v8f cz = {};
        v8f d0 = __builtin_amdgcn_wmma_f32_16x16x4_f32(
            false, A0, false, Bv, (short)0, cz, false, false);
        v8f d1 = __builtin_amdgcn_wmma_f32_16x16x4_f32(
            false, A1, false, Bv, (short)0, cz, false, false);

        int best = 1000;
#pragma unroll
        for (int r = 0; r < 8; ++r) {
            int  k0 = r + hs * 8;        // tile 0 mean index
            int  k1 = 16 + k0;           // tile 1 mean index
            bool m0 = d0[r] < BW2;
            bool m1 = d1[r] < BW2;
            masko[((size_t)(b * K_ + k0)) * P_ + p] = m0 ? 1.0f : 0.0f;
            masko[((size_t)(b * K_ + k1)) * P_ + p] = m1 ? 1.0f : 0.0f;
            best = (m0 && k0 < best) ? k0 : best;
            best = (m1 && k1 < best) ? k1 : best;
        }
        // lanes j and j+16 cover complementary mean subsets for pixel p
        int other = __shfl_xor(best, 16, 32);
        best = best < other ? best : other;
        if (hs == 0)
            labels[(size_t)b * P_ + p] = (best < 1000) ? (float)(best + 1) : 0.0f;
    }
}

// ---------------------------------------------------------------------------
extern "C" void kernel_launch(void* const* d_in, const int* in_sizes, int n_in,
                              void* d_out, int out_size, void* d_ws, size_t ws_size,
                              hipStream_t stream) {
    (void)in_sizes; (void)n_in; (void)out_size; (void)ws_size;

    const float* feat = (const float*)d_in[0];
    const int*   seed = (const int*)d_in[1];
    float*       out  = (float*)d_out;

    float* means    = (float*)d_ws;                  // [B][K][2]
    float* partials = means + B_ * K_ * C_;          // [B][NBLK][K][3] ~96KB

    ms_init_means<<<1, 128, 0, stream>>>(feat, seed, means);
    for (int it = 0; it < NIT; ++it) {
        ms_accum<<<B_ * NBLK, 256, 0, stream>>>(feat, means, partials);
        ms_update<<<1, 128, 0, stream>>>(partials, means);
    }
    ms_write_means<<<1, 256, 0, stream>>>(means, out);
    ms_final<<<B_ * NBLK2, 256, 0, stream>>>(feat, means, out);
}